// CommunicateGCN_78546361909532
// MI455X (gfx1250) — compile-verified
//
#include <hip/hip_runtime.h>

#define N_NODES   10000
#define NUM_GENES 978
#define EMBED_DIM 301
#define HP        320          // padded column stride for Wp / h / out_pre
#define KPAD      992          // 31 * 32, padded K for Wp
#define N_EDGES   320000
#define KB        32           // K-block for GEMM staging
#define KSTEPS    31           // ceil(978 / 32)

typedef float v2f __attribute__((ext_vector_type(2)));
typedef float v8f __attribute__((ext_vector_type(8)));

// ---------------------------------------------------------------- degree prep
__global__ __launch_bounds__(256) void k_init_deg(float* __restrict__ deg) {
  int i = blockIdx.x * 256 + threadIdx.x;
  if (i < N_NODES) deg[i] = 1.0f;                    // self-loop weight = 1
}

__global__ __launch_bounds__(256) void k_deg_accum(const int* __restrict__ dst,
                                                   const float* __restrict__ w,
                                                   float* __restrict__ deg) {
  int e = blockIdx.x * 256 + threadIdx.x;
  if (e < N_EDGES) atomicAdd(&deg[dst[e]], w[e]);
}

__global__ __launch_bounds__(256) void k_rsqrt(float* __restrict__ deg) {
  int i = blockIdx.x * 256 + threadIdx.x;
  if (i < N_NODES) deg[i] = 1.0f / sqrtf(deg[i]);    // deg >= 1 always
}

// ---------------------------------------------------------------- pad W -> Wp
// Wp[KPAD][HP], zero-filled outside [978, 301]: removes ALL guards from GEMM W path
__global__ __launch_bounds__(256) void k_padW(const float* __restrict__ W,
                                              float* __restrict__ Wp) {
  int i = blockIdx.x * 256 + threadIdx.x;
  if (i < KPAD * HP) {
    int k  = i / HP;
    int nn = i - k * HP;
    Wp[i] = (k < NUM_GENES && nn < EMBED_DIM) ? W[k * EMBED_DIM + nn] : 0.f;
  }
}

// ---------------------------------------------------------------- GEMM h = x @ W
// fp32 WMMA 16x16x4.  Block = 8 wave32s, 32(M) x 64(N) tile, register-double-buffered.
__global__ __launch_bounds__(256) void k_gemm(const float* __restrict__ x,
                                              const float* __restrict__ Wp,
                                              float* __restrict__ h) {
  __shared__ float Xs[32][36];       // stride 36: even (b64-aligned), conflict-free reads
  __shared__ float Ws[KB][64];

  const int tid  = threadIdx.x;
  const int wave = tid >> 5;
  const int lane = tid & 31;
  const int half = lane >> 4;        // 0: lanes 0-15, 1: lanes 16-31
  const int l16  = lane & 15;

  const int m0 = blockIdx.x * 32;
  const int n0 = blockIdx.y * 64;
  const int wm = (wave >> 2) * 16;   // 0 or 16
  const int wn = (wave & 3) * 16;    // 0..48

  // X staging: 32x32 tile, 4 floats/thread (2 x b64, always 8B-aligned: 978 even)
  const int xr = tid >> 3;           // 0..31
  const int xc = (tid & 7) * 4;      // 0..28
  const int gm = m0 + xr;
  const bool xrow_ok = gm < N_NODES; // divergent only in the single partial M-block
  const float* xrow = x + (size_t)gm * NUM_GENES;

  // W staging: 32x64 tile, 2 x float4/thread from padded Wp (16B-aligned, no guards)
  const int wr0 = tid >> 4;          // 0..15  (second chunk: +16)
  const int wc0 = (tid & 15) * 4;    // 0..60
  const float* wp0 = Wp + (size_t)wr0 * HP + n0 + wc0;
  const float* wp1 = Wp + (size_t)(wr0 + 16) * HP + n0 + wc0;

  float  xa0, xa1, xa2, xa3;
  float4 wa0, wa1;

  // ---- prefetch tile k0 = 0 (k0+KB <= 978, so X path is unguarded in k)
  if (xrow_ok) {
    float2 t0 = *(const float2*)(xrow + xc);
    float2 t1 = *(const float2*)(xrow + xc + 2);
    xa0 = t0.x; xa1 = t0.y; xa2 = t1.x; xa3 = t1.y;
  } else { xa0 = xa1 = xa2 = xa3 = 0.f; }
  wa0 = *(const float4*)wp0;
  wa1 = *(const float4*)wp1;

  v8f acc = {0.f, 0.f, 0.f, 0.f, 0.f, 0.f, 0.f, 0.f};

  for (int it = 0; it < KSTEPS; ++it) {
    // ---- commit staged registers to LDS
    *(float2*)&Xs[xr][xc]        = make_float2(xa0, xa1);
    *(float2*)&Xs[xr][xc + 2]    = make_float2(xa2, xa3);
    *(float4*)&Ws[wr0][wc0]      = wa0;
    *(float4*)&Ws[wr0 + 16][wc0] = wa1;
    __syncthreads();

    // ---- prefetch next tile (global loads overlap the WMMAs below)
    if (it + 1 < KSTEPS) {
      const int k0 = (it + 1) * KB;
      if (xrow_ok) {
        if (k0 + KB <= NUM_GENES) {
          float2 t0 = *(const float2*)(xrow + k0 + xc);
          float2 t1 = *(const float2*)(xrow + k0 + xc + 2);
          xa0 = t0.x; xa1 = t0.y; xa2 = t1.x; xa3 = t1.y;
        } else {                      // only the last K-block (k0 = 960)
          xa0 = (k0 + xc + 0 < NUM_GENES) ? xrow[k0 + xc + 0] : 0.f;
          xa1 = (k0 + xc + 1 < NUM_GENES) ? xrow[k0 + xc + 1] : 0.f;
          xa2 = (k0 + xc + 2 < NUM_GENES) ? xrow[k0 + xc + 2] : 0.f;
          xa3 = (k0 + xc + 3 < NUM_GENES) ? xrow[k0 + xc + 3] : 0.f;
        }
      } else { xa0 = xa1 = xa2 = xa3 = 0.f; }
      wa0 = *(const float4*)(wp0 + (size_t)k0 * HP);
      wa1 = *(const float4*)(wp1 + (size_t)k0 * HP);
    }

    // ---- 8 f32 WMMAs from LDS
#pragma unroll
    for (int kk = 0; kk < KB; kk += 4) {
      const int ka = kk + half * 2;  // lanes 0-15: K=kk,kk+1 ; lanes 16-31: K=kk+2,kk+3
      v2f a, bfr;
      a.x   = Xs[wm + l16][ka];
      a.y   = Xs[wm + l16][ka + 1];
      bfr.x = Ws[ka][wn + l16];
      bfr.y = Ws[ka + 1][wn + l16];
      acc = __builtin_amdgcn_wmma_f32_16x16x4_f32(
          false, a, false, bfr, (short)0, acc, false, false);
    }
    __syncthreads();
  }

  // ---- epilogue: VGPR r holds M=r (lanes 0-15) / M=8+r (lanes 16-31), N=l16
  const int gn = n0 + wn + l16;      // < HP always; padded cols are exact zeros
#pragma unroll
  for (int r = 0; r < 8; ++r) {
    int gmo = m0 + wm + half * 8 + r;
    if (gmo < N_NODES) h[(size_t)gmo * HP + gn] = acc[r];
  }
}

// ---------------------------------------------------------------- self-loop init
// out_pre[n][d] = dis[n]^2 * h[n][d]   (self loop: norm = dis[n]*1*dis[n])
__global__ __launch_bounds__(320) void k_self(const float* __restrict__ dis,
                                              const float* __restrict__ h,
                                              float* __restrict__ op) {
  const int n = blockIdx.x;
  const float s = dis[n];
  const int i = n * HP + threadIdx.x;
  op[i] = s * s * h[i];
}

// ---------------------------------------------------------------- edge scatter
// one wave32 per edge; lanes stride the 301-dim feature; L2-resident f32 atomics
__global__ __launch_bounds__(256) void k_scatter(const int* __restrict__ src,
                                                 const int* __restrict__ dst,
                                                 const float* __restrict__ w,
                                                 const float* __restrict__ dis,
                                                 const float* __restrict__ h,
                                                 float* __restrict__ op) {
  int gwave = (blockIdx.x * 256 + threadIdx.x) >> 5;   // edge id (wave-uniform)
  int lane  = threadIdx.x & 31;
  if (gwave >= N_EDGES) return;
  int s = src[gwave];
  int t = dst[gwave];
  float coeff = dis[s] * w[gwave] * dis[t];
  const float* hs = h + (size_t)s * HP;
  float* ot       = op + (size_t)t * HP;
  for (int d = lane; d < EMBED_DIM; d += 32)
    atomicAdd(&ot[d], coeff * hs[d]);
}

// ---------------------------------------------------------------- epilogue
// out[n, hd, d, r] = relu(out_pre[n,d] + b[d]), broadcast over hd(4) x r(12)
// 578 MB of stores = the roofline floor; 3 x b128 stores per 12-float run, no int div.
__global__ __launch_bounds__(256) void k_final(const float* __restrict__ op,
                                               const float* __restrict__ b,
                                               float* __restrict__ out) {
  __shared__ float sv[EMBED_DIM];
  const int n = blockIdx.x;
  for (int d = threadIdx.x; d < EMBED_DIM; d += 256) {
    float v = op[(size_t)n * HP + d] + b[d];
    sv[d] = v > 0.f ? v : 0.f;
  }
  __syncthreads();
  float* o = out + (size_t)n * (4 * EMBED_DIM * 12);
#pragma unroll
  for (int hd = 0; hd < 4; ++hd) {
    float* oh = o + hd * (EMBED_DIM * 12);
    for (int d = threadIdx.x; d < EMBED_DIM; d += 256) {
      const float  v  = sv[d];
      const float4 vv = make_float4(v, v, v, v);
      float4* p = (float4*)(oh + d * 12);       // d*12 floats: 16B aligned
      p[0] = vv; p[1] = vv; p[2] = vv;
    }
  }
}

// ---------------------------------------------------------------- launcher
extern "C" void kernel_launch(void* const* d_in, const int* in_sizes, int n_in,
                              void* d_out, int out_size, void* d_ws, size_t ws_size,
                              hipStream_t stream) {
  (void)in_sizes; (void)n_in; (void)out_size; (void)ws_size;

  const float* x  = (const float*)d_in[0];
  const int*   ei = (const int*)d_in[1];     // [2, E] int32 (JAX x64 off)
  const float* ew = (const float*)d_in[2];
  const float* W  = (const float*)d_in[3];
  const float* b  = (const float*)d_in[4];
  float* out = (float*)d_out;

  const int* src = ei;
  const int* dst = ei + N_EDGES;

  // workspace: deg/dis [10000] | Wp [992 x 320] | h [10000 x 320] | out_pre [10000 x 320]
  float* deg = (float*)d_ws;
  float* Wp  = (float*)((char*)d_ws + 65536);
  float* h   = Wp + (size_t)KPAD * HP;
  float* op  = h + (size_t)N_NODES * HP;

  k_init_deg<<<(N_NODES + 255) / 256, 256, 0, stream>>>(deg);
  k_deg_accum<<<(N_EDGES + 255) / 256, 256, 0, stream>>>(dst, ew, deg);
  k_rsqrt<<<(N_NODES + 255) / 256, 256, 0, stream>>>(deg);
  k_padW<<<(KPAD * HP + 255) / 256, 256, 0, stream>>>(W, Wp);
  k_gemm<<<dim3((N_NODES + 31) / 32, HP / 64), 256, 0, stream>>>(x, Wp, h);
  k_self<<<N_NODES, 320, 0, stream>>>(deg, h, op);
  k_scatter<<<N_EDGES / 8, 256, 0, stream>>>(src, dst, ew, deg, h, op);
  k_final<<<N_NODES, 256, 0, stream>>>(op, b, out);
}